// NeighborAttention_16827681866239
// MI455X (gfx1250) — compile-verified
//
#include <hip/hip_runtime.h>
#include <math.h>

// ---- problem constants (from reference) ----
#define BB   4
#define NN   2048
#define KK   30
#define HH   8
#define DD   64
#define CIN  640      // NUM_IN
#define NHID 512      // NUM_HIDDEN
#define NLAT 128
#define NBW  144      // per-head fused weight panel: 64 Kp | 64 V | 1 bias | 15 pad
#define NEGMAX (-3.402823466e+38f)

typedef float v2f __attribute__((ext_vector_type(2)));
typedef float v8f __attribute__((ext_vector_type(8)));

static __device__ __forceinline__ v8f wmma_f32(v2f a, v2f b, v8f c) {
  return __builtin_amdgcn_wmma_f32_16x16x4_f32(false, a, false, b, (short)0, c, false, false);
}

// ---------------- prep: fused per-head weight panel, K-FAST: Wcat_t[h][144][640] ----------------
__global__ void prep_wcat(const float* __restrict__ Wk, const float* __restrict__ Wv,
                          const float* __restrict__ Wb, float* __restrict__ Wcat) {
  int idx = blockIdx.x * 256 + threadIdx.x;       // idx = (h*NBW + nloc)*CIN + r
  if (idx >= HH * NBW * CIN) return;
  int r    = idx % CIN;
  int nloc = (idx / CIN) % NBW;
  int h    = idx / (CIN * NBW);
  float v = 0.f;
  if (nloc < 64)        v = Wk[(size_t)r * NHID + h * 64 + nloc];
  else if (nloc < 128)  v = Wv[(size_t)r * NHID + h * 64 + (nloc - 64)];
  else if (nloc == 128) v = (r >= 33) ? Wb[(size_t)(r - 33) * HH + h] : 0.f;
  Wcat[idx] = v;
}

// ---------------- prep: transpose KxN -> out[n][k] (for W_Q / W_O) ----------------
__global__ void transpose_nk(const float* __restrict__ in, float* __restrict__ out, int K, int N) {
  int idx = blockIdx.x * 256 + threadIdx.x;       // idx = n*K + k
  if (idx >= N * K) return;
  int k = idx % K, n = idx / K;
  out[idx] = in[(size_t)k * N + n];
}

// ---------------- prep: bias_latent[B][H][K] = (z @ W_z) permuted ----------------
__global__ void prep_blat(const float* __restrict__ z, const float* __restrict__ Wz,
                          float* __restrict__ blat) {
  int idx = blockIdx.x * 256 + threadIdx.x;
  if (idx >= BB * HH * KK) return;
  int k = idx % KK, h = (idx / KK) % HH, b = idx / (KK * HH);
  float s = 0.f;
  for (int l = 0; l < NLAT; ++l)
    s += z[b * NLAT + l] * Wz[l * (KK * HH) + k * HH + h];
  blat[idx] = s;
}

// ---------------- generic f32 WMMA GEMM: C(MxN) = A(MxK) @ Bt(NxK)^T, tiles 128x64 ----------------
// Bt is the K-fast (transposed) weight panel. Double-buffered LDS, k-fast fragments.
__global__ __launch_bounds__(256) void gemm_f32_wmma(const float* __restrict__ A,
                                                     const float* __restrict__ Bt,
                                                     float* __restrict__ C,
                                                     int M, int N, int K) {
  __shared__ float As[2][128][36];   // rows x 32-wide K chunk (stride 36, 16B aligned)
  __shared__ float Bs[2][64][36];    // N-rows x 32-wide K chunk (k-fast!)
  const int tid  = threadIdx.x;
  const int wave = tid >> 5, lane = tid & 31;
  const int col  = lane & 15, half = lane >> 4, ko = half * 2;
  const int m0 = blockIdx.y * 128, n0 = blockIdx.x * 64;
  const int NC = K >> 5;

  v8f acc[4] = {};

  // prologue: stage chunk 0 into buffer 0
  {
    float4 ra[4], rb[2];
#pragma unroll
    for (int i = 0; i < 4; ++i) {
      int s = tid + i * 256, r = s >> 3, c4 = (s & 7) * 4;
      ra[i] = *(const float4*)&A[(size_t)(m0 + r) * K + c4];
    }
#pragma unroll
    for (int i = 0; i < 2; ++i) {
      int s = tid + i * 256, n = s >> 3, c4 = (s & 7) * 4;
      rb[i] = *(const float4*)&Bt[(size_t)(n0 + n) * K + c4];
    }
#pragma unroll
    for (int i = 0; i < 4; ++i) {
      int s = tid + i * 256, r = s >> 3, c4 = (s & 7) * 4;
      *(float4*)&As[0][r][c4] = ra[i];
    }
#pragma unroll
    for (int i = 0; i < 2; ++i) {
      int s = tid + i * 256, n = s >> 3, c4 = (s & 7) * 4;
      *(float4*)&Bs[0][n][c4] = rb[i];
    }
  }

  for (int c = 0; c < NC; ++c) {
    const int cur = c & 1;
    const bool more = (c + 1) < NC;
    float4 na[4], nb[2];
    if (more) {                                   // issue next chunk's global loads now
      int kc = (c + 1) << 5;
#pragma unroll
      for (int i = 0; i < 4; ++i) {
        int s = tid + i * 256, r = s >> 3, c4 = (s & 7) * 4;
        na[i] = *(const float4*)&A[(size_t)(m0 + r) * K + kc + c4];
      }
#pragma unroll
      for (int i = 0; i < 2; ++i) {
        int s = tid + i * 256, n = s >> 3, c4 = (s & 7) * 4;
        nb[i] = *(const float4*)&Bt[(size_t)(n0 + n) * K + kc + c4];
      }
    }
    __syncthreads();                              // buf[cur] visible to all waves
#pragma unroll
    for (int ks = 0; ks < 32; ks += 4) {
      v2f a = *(const v2f*)&As[cur][wave * 16 + col][ks + ko];
#pragma unroll
      for (int nt = 0; nt < 4; ++nt) {
        v2f b = *(const v2f*)&Bs[cur][nt * 16 + col][ks + ko];
        acc[nt] = wmma_f32(a, b, acc[nt]);
      }
    }
    if (more) {
#pragma unroll
      for (int i = 0; i < 4; ++i) {
        int s = tid + i * 256, r = s >> 3, c4 = (s & 7) * 4;
        *(float4*)&As[cur ^ 1][r][c4] = na[i];
      }
#pragma unroll
      for (int i = 0; i < 2; ++i) {
        int s = tid + i * 256, n = s >> 3, c4 = (s & 7) * 4;
        *(float4*)&Bs[cur ^ 1][n][c4] = nb[i];
      }
    }
  }

#pragma unroll
  for (int nt = 0; nt < 4; ++nt)
#pragma unroll
    for (int m = 0; m < 8; ++m) {
      int r = m0 + wave * 16 + m + half * 8;
      C[(size_t)r * N + n0 + nt * 16 + col] = acc[nt][m];
    }
}

// ---------------- fused edge kernel ----------------
// grid = (B*N/8, H), block = 256 (8 waves). Wave w owns (b,n)-pair w of the group
// (32 padded edge rows = 2 M-tiles) x one head's 144-col fused panel (9 N-tiles).
// Dynamic LDS (~116 KB of the 320 KB WGP budget), double-buffered, k-fast fragments.
#define A_BUF (256 * 36)
#define B_BUF (144 * 36)
__global__ __launch_bounds__(256) void edge_attn(const float* __restrict__ hE,
                                                 const float* __restrict__ Qws,
                                                 const float* __restrict__ Wcat,   // k-fast [h][144][640]
                                                 const float* __restrict__ blat,
                                                 const float* __restrict__ maskA,
                                                 float* __restrict__ hupd) {
  extern __shared__ float smem[];
  // layout: A buf0 | A buf1 | B buf0 | B buf1 | logits[8][32]
  float* logits_lds = smem + 2 * A_BUF + 2 * B_BUF;

  const int tid = threadIdx.x;
  const int h = blockIdx.y;
  const int bn0 = blockIdx.x * 8;
  const int wave = tid >> 5, lane = tid & 31;
  const int col = lane & 15, half = lane >> 4, ko = half * 2;
  const int NC = CIN >> 5;                             // 20 chunks

  v8f acc[2][9] = {};

  // prologue: stage chunk 0 into buffer 0
  {
    float4 ra[8], rb[5];
#pragma unroll
    for (int i = 0; i < 8; ++i) {
      int s = tid + i * 256, r = s >> 3, c4 = (s & 7) * 4;
      int p = r >> 5, e = r & 31;
      ra[i] = make_float4(0.f, 0.f, 0.f, 0.f);
      if (e < KK) ra[i] = *(const float4*)&hE[((size_t)(bn0 + p) * KK + e) * CIN + c4];
    }
#pragma unroll
    for (int i = 0; i < 5; ++i) {
      int s = tid + i * 256;
      if (s < 1152) {
        int n = s >> 3, c4 = (s & 7) * 4;
        rb[i] = *(const float4*)&Wcat[((size_t)h * NBW + n) * CIN + c4];
      }
    }
#pragma unroll
    for (int i = 0; i < 8; ++i) {
      int s = tid + i * 256, r = s >> 3, c4 = (s & 7) * 4;
      *(float4*)&smem[r * 36 + c4] = ra[i];
    }
#pragma unroll
    for (int i = 0; i < 5; ++i) {
      int s = tid + i * 256;
      if (s < 1152) {
        int n = s >> 3, c4 = (s & 7) * 4;
        *(float4*)&smem[2 * A_BUF + n * 36 + c4] = rb[i];
      }
    }
  }

  for (int c = 0; c < NC; ++c) {
    const int cur = c & 1;
    const bool more = (c + 1) < NC;
    float4 na[8], nb[5];
    if (more) {                                   // issue next chunk's global loads
      int kc = (c + 1) << 5;
#pragma unroll
      for (int i = 0; i < 8; ++i) {
        int s = tid + i * 256, r = s >> 3, c4 = (s & 7) * 4;
        int p = r >> 5, e = r & 31;
        na[i] = make_float4(0.f, 0.f, 0.f, 0.f);
        if (e < KK) na[i] = *(const float4*)&hE[((size_t)(bn0 + p) * KK + e) * CIN + kc + c4];
      }
#pragma unroll
      for (int i = 0; i < 5; ++i) {
        int s = tid + i * 256;
        if (s < 1152) {
          int n = s >> 3, c4 = (s & 7) * 4;
          nb[i] = *(const float4*)&Wcat[((size_t)h * NBW + n) * CIN + kc + c4];
        }
      }
    }
    __syncthreads();
    const float* Ac = smem + cur * A_BUF;
    const float* Bc = smem + 2 * A_BUF + cur * B_BUF;
#pragma unroll
    for (int ks = 0; ks < 32; ks += 4) {
      v2f a0 = *(const v2f*)&Ac[(wave * 32 + col) * 36 + ks + ko];
      v2f a1 = *(const v2f*)&Ac[(wave * 32 + 16 + col) * 36 + ks + ko];
#pragma unroll
      for (int nt = 0; nt < 9; ++nt) {
        v2f b = *(const v2f*)&Bc[(nt * 16 + col) * 36 + ks + ko];
        acc[0][nt] = wmma_f32(a0, b, acc[0][nt]);
        acc[1][nt] = wmma_f32(a1, b, acc[1][nt]);
      }
    }
    if (more) {
      float* An = smem + (cur ^ 1) * A_BUF;
      float* Bn = smem + 2 * A_BUF + (cur ^ 1) * B_BUF;
#pragma unroll
      for (int i = 0; i < 8; ++i) {
        int s = tid + i * 256, r = s >> 3, c4 = (s & 7) * 4;
        *(float4*)&An[r * 36 + c4] = na[i];
      }
#pragma unroll
      for (int i = 0; i < 5; ++i) {
        int s = tid + i * 256;
        if (s < 1152) {
          int n = s >> 3, c4 = (s & 7) * 4;
          *(float4*)&Bn[n * 36 + c4] = nb[i];
        }
      }
    }
  }

  // ---- epilogue: logits -> softmax -> attend@V, all within wave `wave` ----
  const int bn = bn0 + wave;
  const int b  = bn / NN;

  float q[4];
#pragma unroll
  for (int nt = 0; nt < 4; ++nt)
    q[nt] = Qws[(size_t)bn * NHID + h * 64 + nt * 16 + col];

#pragma unroll
  for (int t = 0; t < 2; ++t)
#pragma unroll
    for (int m = 0; m < 8; ++m) {
      float lp = 0.f;
#pragma unroll
      for (int nt = 0; nt < 4; ++nt) lp += q[nt] * acc[t][nt][m];
      lp += __shfl_xor(lp, 1, 32);
      lp += __shfl_xor(lp, 2, 32);
      lp += __shfl_xor(lp, 4, 32);
      lp += __shfl_xor(lp, 8, 32);               // row-sum within each 16-lane half
      int kr = t * 16 + m + half * 8;            // edge index for this half
      float be = __shfl(acc[t][8][m], half * 16, 32);   // bias_edges (panel col 128)
      float bl = (kr < KK) ? blat[(b * HH + h) * KK + kr] : 0.f;
      float mk = (kr < KK) ? maskA[(size_t)bn * KK + kr] : 0.f;
      float lg = lp * 0.125f + be + bl;          // 1/sqrt(64)
      lg = (mk > 0.f) ? lg : NEGMAX;
      if (lane == half * 16) logits_lds[wave * 32 + kr] = lg;
    }
  __syncthreads();

  float lv = (lane < KK) ? logits_lds[wave * 32 + lane] : NEGMAX;
  float mx = lv;
  mx = fmaxf(mx, __shfl_xor(mx, 1, 32));
  mx = fmaxf(mx, __shfl_xor(mx, 2, 32));
  mx = fmaxf(mx, __shfl_xor(mx, 4, 32));
  mx = fmaxf(mx, __shfl_xor(mx, 8, 32));
  mx = fmaxf(mx, __shfl_xor(mx, 16, 32));
  float ex = __expf(lv - mx);
  if (lane >= KK) ex = 0.f;
  float sm = ex;
  sm += __shfl_xor(sm, 1, 32);
  sm += __shfl_xor(sm, 2, 32);
  sm += __shfl_xor(sm, 4, 32);
  sm += __shfl_xor(sm, 8, 32);
  sm += __shfl_xor(sm, 16, 32);
  float at = ex / sm;
  at *= (lane < KK) ? maskA[(size_t)bn * KK + lane] : 0.f;   // attend * mask

#pragma unroll
  for (int nt = 0; nt < 4; ++nt) {               // h_upd[d] = sum_k attend[k]*V[k,d]
    float s = 0.f;
#pragma unroll
    for (int t = 0; t < 2; ++t)
#pragma unroll
      for (int m = 0; m < 8; ++m) {
        float am = __shfl(at, t * 16 + m + half * 8, 32);
        s += am * acc[t][4 + nt][m];
      }
    s += __shfl_xor(s, 16, 32);                  // combine row halves
    if (half == 0)
      hupd[(size_t)bn * NHID + h * 64 + nt * 16 + col] = s;
  }
}

// ---------------- launch ----------------
extern "C" void kernel_launch(void* const* d_in, const int* in_sizes, int n_in,
                              void* d_out, int out_size, void* d_ws, size_t ws_size,
                              hipStream_t stream) {
  const float* z     = (const float*)d_in[0];
  const float* h_V   = (const float*)d_in[1];
  const float* h_E   = (const float*)d_in[2];
  const float* maskA = (const float*)d_in[3];
  const float* W_Q   = (const float*)d_in[4];
  const float* W_K   = (const float*)d_in[5];
  const float* W_V   = (const float*)d_in[6];
  const float* W_O   = (const float*)d_in[7];
  const float* W_z   = (const float*)d_in[8];
  const float* W_b   = (const float*)d_in[9];

  const int M = BB * NN;                // 8192 rows
  float* ws   = (float*)d_ws;
  float* Qws  = ws;                                    // M*512
  float* hupd = Qws  + (size_t)M * NHID;               // M*512
  float* Wcat = hupd + (size_t)M * NHID;               // H*144*640 (k-fast)
  float* blat = Wcat + (size_t)HH * NBW * CIN;         // B*H*K
  float* WQt  = blat + (size_t)BB * HH * KK;           // 512*512 transposed
  float* WOt  = WQt  + (size_t)NHID * NHID;            // 512*512 transposed

  prep_wcat<<<(HH * NBW * CIN + 255) / 256, 256, 0, stream>>>(W_K, W_V, W_b, Wcat);
  transpose_nk<<<(NHID * NHID + 255) / 256, 256, 0, stream>>>(W_Q, WQt, NHID, NHID);
  transpose_nk<<<(NHID * NHID + 255) / 256, 256, 0, stream>>>(W_O, WOt, NHID, NHID);
  prep_blat<<<(BB * HH * KK + 255) / 256, 256, 0, stream>>>(z, W_z, blat);

  // Q = h_V @ W_Q
  gemm_f32_wmma<<<dim3(NHID / 64, M / 128), 256, 0, stream>>>(h_V, WQt, Qws, M, NHID, NHID);

  // fused K/V/bias projection + attention
  size_t shmem = (2 * A_BUF + 2 * B_BUF + 8 * 32) * sizeof(float);
  edge_attn<<<dim3(M / 8, HH), 256, shmem, stream>>>(h_E, Qws, Wcat, blat, maskA, hupd);

  // out = h_upd @ W_O
  gemm_f32_wmma<<<dim3(NHID / 64, M / 128), 256, 0, stream>>>(hupd, WOt, (float*)d_out, M, NHID, NHID);
}